// RelationalEmbedding_41549513622290
// MI455X (gfx1250) — compile-verified
//
#include <hip/hip_runtime.h>

// RelationalEmbedding, algebraically collapsed (the two affine layers have no
// nonlinearity, so the per-pair MLP is a single dot with v = W1@W2, c = b1.W2+b2,
// and the axis-0 sum factors into out[j] = seq[j] * s[j]).
// Dominant cost: streaming rel_encoding (134 MB) once per hop -> bandwidth bound.

#define N_   1024
#define U_   64
#define K_   32
#define IN_  160

typedef __attribute__((ext_vector_type(2))) float v2f;
typedef __attribute__((ext_vector_type(8))) float v8f;

// workspace layout (float offsets)
#define WS_V1    0        // 160
#define WS_V2    160      // 160
#define WS_C     320      // c1, c2
#define WS_INVD  512      // 1024
#define WS_ACD   1536     // 1024  (a[i]+c)*invD[i]
#define WS_BB    2560     // 1024  seq[j].vB
#define WS_S     3584     // 1024
#define WS_SEQM  4608     // 65536 hop-1 output
#define WS_SPART 70144    // 32*1024 partial column sums
// total = 102912 floats = ~402 KB

#define JT_     32                 // j's per block in streaming kernel
#define ICH_    32                 // i's per block
#define JTILES_ (N_ / JT_)         // 32
#define ITILES_ (N_ / ICH_)        // 32

// ---------------------------------------------------------------------------
// Kernel 1: collapse weights.  Blocks 0-19: v_h = W_h1 @ W_h2 via WMMA f32
// 16x16x4 (10 row tiles per hop).  Blocks 20-21: c_h = b_h1.W_h2 + b_h2.
// One wave per block; EXEC is all-ones before any WMMA (uniform branch).
__global__ __launch_bounds__(32)
void k_collapse(const float* __restrict__ W11, const float* __restrict__ b11,
                const float* __restrict__ W12, const float* __restrict__ b12,
                const float* __restrict__ W21, const float* __restrict__ b21,
                const float* __restrict__ W22, const float* __restrict__ b22,
                float* __restrict__ ws)
{
    const int bid  = blockIdx.x;
    const int lane = threadIdx.x;
    if (bid < 20) {
        const int hop  = bid / 10;
        const int tile = bid % 10;
        const float* W1 = hop ? W21 : W11;
        const float* w2 = hop ? W22 : W12;          // (IN,1) column
        float* vout = ws + (hop ? WS_V2 : WS_V1);
        const int p0   = tile * 16;
        const int half = lane >> 4;                 // K slot pair selector
        const int m    = lane & 15;                 // row within tile
        v8f acc = {};
        for (int k0 = 0; k0 < IN_; k0 += 4) {
            const int ka = k0 + half * 2;
            v2f A, B;
            A.x = W1[(p0 + m) * IN_ + ka];
            A.y = W1[(p0 + m) * IN_ + ka + 1];
            // replicate w2 across all 16 N-columns: result invariant to the
            // column labeling and to any consistent K-slot permutation.
            B.x = w2[ka];
            B.y = w2[ka + 1];
            acc = __builtin_amdgcn_wmma_f32_16x16x4_f32(
                false, A, false, B, (short)0, acc, false, false);
        }
        // D layout: VGPR r -> M=r (lanes 0-15) / M=8+r (lanes 16-31); all
        // columns identical, so lanes 0 and 16 carry rows 0-7 / 8-15.
        if (lane == 0) {
            #pragma unroll
            for (int r = 0; r < 8; ++r) vout[p0 + r] = acc[r];
        }
        if (lane == 16) {
            #pragma unroll
            for (int r = 0; r < 8; ++r) vout[p0 + 8 + r] = acc[r];
        }
    } else {
        const int hop = bid - 20;
        const float* b1 = hop ? b21 : b11;
        const float* w2 = hop ? W22 : W12;
        const float* b2 = hop ? b22 : b12;
        float val = 0.f;
        for (int q = lane; q < IN_; q += 32) val += b1[q] * w2[q];
        #pragma unroll
        for (int off = 16; off >= 1; off >>= 1) val += __shfl_xor(val, off, 32);
        if (lane == 0) ws[WS_C + hop] = val + b2[0];
    }
}

// ---------------------------------------------------------------------------
// Kernel 2: invD[i] = 1 / sum_j mask[i,j]   (D >= 1: identity included)
__global__ __launch_bounds__(256)
void k_invd(const float* __restrict__ mask, float* __restrict__ invD)
{
    __shared__ float red[256];
    const int i = blockIdx.x, t = threadIdx.x;
    float s = 0.f;
    for (int j = t; j < N_; j += 256) s += mask[(size_t)i * N_ + j];
    red[t] = s;
    __syncthreads();
    for (int off = 128; off >= 32; off >>= 1) {
        if (t < off) red[t] += red[t + off];
        __syncthreads();
    }
    if (t < 32) {
        float v = red[t];
        #pragma unroll
        for (int off = 16; off >= 1; off >>= 1) v += __shfl_xor(v, off, 32);
        if (t == 0) invD[i] = 1.0f / v;
    }
}

// ---------------------------------------------------------------------------
// Kernel 3: per-row dots.  acd[i] = (seq[i].vA + c) * invD[i], bb[i] = seq[i].vB
__global__ __launch_bounds__(64)
void k_rowdots(const float* __restrict__ seq, const float* __restrict__ v,
               const float* __restrict__ cptr, const float* __restrict__ invD,
               float* __restrict__ acd, float* __restrict__ bb)
{
    __shared__ float sa[64], sb[64];
    const int i = blockIdx.x, t = threadIdx.x;
    const float e = seq[(size_t)i * U_ + t];
    sa[t] = e * v[t];
    sb[t] = e * v[U_ + t];
    __syncthreads();
    if (t < 32) {
        float A = sa[t] + sa[t + 32];
        float B = sb[t] + sb[t + 32];
        #pragma unroll
        for (int off = 16; off >= 1; off >>= 1) {
            A += __shfl_xor(A, off, 32);
            B += __shfl_xor(B, off, 32);
        }
        if (t == 0) {
            acd[i] = (A + cptr[0]) * invD[i];
            bb[i]  = B;
        }
    }
}

// ---------------------------------------------------------------------------
// Kernel 4: streaming kernel over rel_encoding (the bandwidth-bound core).
// Thread t: j-group g=t/8 (32 j's/block), k-group l=t%8 (float4 of K).
// A wave's 32 lanes read 512 contiguous bytes of rel per i (fully coalesced).
// Per lane per i: 4 FMA (acc += mi*rel4) + mi/msum/asum bookkeeping; the
// k-reduction happens ONCE at the end (3 xor-shuffles within groups of 8).
// Writes deterministic partials: spart[it*N + j].
__global__ __launch_bounds__(256)
void k_stream(const float* __restrict__ rel, const float* __restrict__ mask,
              const float* __restrict__ v,  const float* __restrict__ acd,
              const float* __restrict__ invD, const float* __restrict__ bb,
              float* __restrict__ spart)
{
    const int t  = threadIdx.x;
    const int g  = t >> 3;
    const int l  = t & 7;
    const int k4 = l * 4;
    const int jt = blockIdx.x % JTILES_;
    const int it = blockIdx.x / JTILES_;
    const int j  = jt * JT_ + g;
    const int i0 = it * ICH_;

    const float4 vc4 = *(const float4*)(v + 2 * U_ + k4);

    float4 acc = make_float4(0.f, 0.f, 0.f, 0.f);
    float msum = 0.f, asum = 0.f;

    const float* relp = rel + ((size_t)i0 * N_ + j) * K_ + k4;
    const size_t istride = (size_t)N_ * K_;

    for (int ii = 0; ii < ICH_; ++ii) {
        if (ii + 4 < ICH_)                       // in-bounds lookahead
            __builtin_prefetch(relp + 4 * istride, 0, 0);
        const float4 r4 = *(const float4*)relp;  // global_load_b128
        const int i = i0 + ii;
        const float mk = mask[(size_t)i * N_ + j];   // shared line across group
        const float mi = mk * invD[i];
        acc.x = fmaf(mi, r4.x, acc.x);
        acc.y = fmaf(mi, r4.y, acc.y);
        acc.z = fmaf(mi, r4.z, acc.z);
        acc.w = fmaf(mi, r4.w, acc.w);
        msum += mi;
        asum  = fmaf(mk, acd[i], asum);          // = mi * (a[i]+c)
        relp += istride;
    }

    float rdot = acc.x * vc4.x + acc.y * vc4.y + acc.z * vc4.z + acc.w * vc4.w;
    rdot += __shfl_xor(rdot, 1, 32);
    rdot += __shfl_xor(rdot, 2, 32);
    rdot += __shfl_xor(rdot, 4, 32);

    if (l == 0)
        spart[(size_t)it * N_ + j] = asum + bb[j] * msum + rdot;
}

// ---------------------------------------------------------------------------
// Kernel 4b: deterministic reduction of the 32 i-chunk partials per column.
__global__ __launch_bounds__(256)
void k_reduce_s(const float* __restrict__ spart, float* __restrict__ s)
{
    const int j = blockIdx.x * 256 + threadIdx.x;
    float acc = 0.f;
    #pragma unroll 8
    for (int it = 0; it < ITILES_; ++it) acc += spart[(size_t)it * N_ + j];
    s[j] = acc;
}

// ---------------------------------------------------------------------------
// Kernel 5: out[j,:] = s[j] * seq_in[j,:]
__global__ __launch_bounds__(256)
void k_scale(const float* __restrict__ seq_in, const float* __restrict__ s,
             float* __restrict__ out)
{
    const int idx = blockIdx.x * 256 + threadIdx.x;
    out[idx] = s[idx >> 6] * seq_in[idx];
}

// ---------------------------------------------------------------------------
extern "C" void kernel_launch(void* const* d_in, const int* in_sizes, int n_in,
                              void* d_out, int out_size, void* d_ws, size_t ws_size,
                              hipStream_t stream)
{
    const float* seq  = (const float*)d_in[0];
    const float* rel  = (const float*)d_in[1];
    const float* mask = (const float*)d_in[2];
    const float* W11  = (const float*)d_in[3];
    const float* b11  = (const float*)d_in[4];
    const float* W12  = (const float*)d_in[5];
    const float* b12  = (const float*)d_in[6];
    const float* W21  = (const float*)d_in[7];
    const float* b21  = (const float*)d_in[8];
    const float* W22  = (const float*)d_in[9];
    const float* b22  = (const float*)d_in[10];
    float* out = (float*)d_out;
    float* ws  = (float*)d_ws;

    float* v1    = ws + WS_V1;
    float* v2    = ws + WS_V2;
    float* c1    = ws + WS_C;
    float* c2    = ws + WS_C + 1;
    float* invD  = ws + WS_INVD;
    float* acd   = ws + WS_ACD;
    float* bb    = ws + WS_BB;
    float* s     = ws + WS_S;
    float* seqm  = ws + WS_SEQM;
    float* spart = ws + WS_SPART;

    (void)in_sizes; (void)n_in; (void)out_size; (void)ws_size;

    k_collapse<<<22, 32, 0, stream>>>(W11, b11, W12, b12, W21, b21, W22, b22, ws);
    k_invd<<<N_, 256, 0, stream>>>(mask, invD);

    // hop 1
    k_rowdots<<<N_, 64, 0, stream>>>(seq, v1, c1, invD, acd, bb);
    k_stream<<<JTILES_ * ITILES_, 256, 0, stream>>>(rel, mask, v1, acd, invD, bb, spart);
    k_reduce_s<<<N_ / 256, 256, 0, stream>>>(spart, s);
    k_scale<<<(N_ * U_) / 256, 256, 0, stream>>>(seq, s, seqm);

    // hop 2 (rel likely resident in the 192 MB L2 now)
    k_rowdots<<<N_, 64, 0, stream>>>(seqm, v2, c2, invD, acd, bb);
    k_stream<<<JTILES_ * ITILES_, 256, 0, stream>>>(rel, mask, v2, acd, invD, bb, spart);
    k_reduce_s<<<N_ / 256, 256, 0, stream>>>(spart, s);
    k_scale<<<(N_ * U_) / 256, 256, 0, stream>>>(seqm, s, out);
}